// Encoder_609885356303
// MI455X (gfx1250) — compile-verified
//
#include <hip/hip_runtime.h>
#include <hip/hip_bf16.h>

// ---------------------------------------------------------------------------
// 3-layer GraphSAGE (mean aggregation) + PReLU for MI455X (gfx1250, wave32).
// f32 WMMA (V_WMMA_F32_16X16X4_F32) for the dense layers; f32 atomics for the
// segment-mean scatter. Memory/atomic bound workload -> keep full precision.
// GEMM blocks are persistent over row strips with weight (B) fragments cached
// in registers, so the steady-state loop is ds_load_b64 + v_wmma only.
// ---------------------------------------------------------------------------

#define N0_ 400000
#define N1_ 200000
#define N2_ 100000
#define N3_ 50000
#define E1_ 500000
#define E2_ 300000
#define E3_ 150000
#define D_  128

typedef __attribute__((ext_vector_type(2))) float v2f;
typedef __attribute__((ext_vector_type(8))) float v8f;

#define LDS_STRIDE 132   // 128 + 4 pad: row*132 % 64 banks spreads across rows

// ---------------------------------------------------------------------------
__global__ __launch_bounds__(256) void zero_f32(float* __restrict__ p, long n)
{
    long i = (long)blockIdx.x * 256 + threadIdx.x;
    if (i < n) p[i] = 0.0f;
}

// One wave32 per edge: lane moves 4 floats (float4 load, 4 f32 atomics).
__global__ __launch_bounds__(256) void scatter_accum(
    const float* __restrict__ x,
    const int*   __restrict__ src,
    const int*   __restrict__ dst,
    float*       __restrict__ sum,
    float*       __restrict__ cnt,
    int E)
{
    int gid  = blockIdx.x * 256 + threadIdx.x;
    int e    = gid >> 5;
    int lane = gid & 31;
    if (e >= E) return;

    int s = src[e];
    int d = dst[e];

    const float4 v = *(const float4*)(x + (size_t)s * D_ + lane * 4);
    float* p = sum + (size_t)d * D_ + lane * 4;
    atomicAdd(p + 0, v.x);
    atomicAdd(p + 1, v.y);
    atomicAdd(p + 2, v.z);
    atomicAdd(p + 3, v.w);
    if (lane == 0) atomicAdd(cnt + d, 1.0f);
}

// ---------------------------------------------------------------------------
// out[m,n] = PReLU( mean[m,:] @ Wl[:,n] + xdst[m,:] @ Wr[:,n] + b[n] )
// Block = 256 threads = 8 waves, persistent over 16-row strips (grid-stride).
// Each wave owns one 16-column N-tile; its B fragments (4x16 f32 per k-step,
// 32 steps x 2 weight matrices = 64 v2f regs) are loaded ONCE per block and
// reused for every strip. A fragments (16x4 f32) come from padded LDS.
// ISA layouts (16x16x4 f32):
//   A: lane L -> M = L%16, VGPR v -> K = 2*(L/16) + v
//   B: lane L -> N = L%16, VGPR v -> K = 2*(L/16) + v
//   C: lane L -> N = L%16, VGPR v -> M = v + 8*(L/16)
// ---------------------------------------------------------------------------
__global__ __launch_bounds__(256) void sage_gemm_prelu(
    const float* __restrict__ xdst,   // >= n_dst x 128 (first n_dst rows used)
    const float* __restrict__ sum,    // n_dst x 128 (segment sums)
    const float* __restrict__ cnt,    // n_dst (degrees)
    const float* __restrict__ Wl,     // 128 x 128 row-major
    const float* __restrict__ Wr,     // 128 x 128 row-major
    const float* __restrict__ bias,   // 128
    const float* __restrict__ alpha,  // 128
    float*       __restrict__ out,    // n_dst x 128
    int n_dst)
{
    __shared__ float ldsMean[16 * LDS_STRIDE];
    __shared__ float ldsX[16 * LDS_STRIDE];

    const int t    = threadIdx.x;
    const int wave = t >> 5;        // 0..7 -> N tile
    const int lane = t & 31;
    const int mn   = lane & 15;     // M for A-frag, N for B/C-frag
    const int half = lane >> 4;     // 0/1
    const int n    = wave * 16 + mn;

    // ---- Preload this wave's weight fragments into registers (once). ----
    v2f bl[32], br[32];
    {
        const float* bLp = Wl + (size_t)(2 * half) * D_ + n;
        const float* bRp = Wr + (size_t)(2 * half) * D_ + n;
#pragma unroll
        for (int kk = 0; kk < 32; ++kk) {
            bl[kk].x = bLp[(size_t)(4 * kk) * D_];
            bl[kk].y = bLp[(size_t)(4 * kk + 1) * D_];
            br[kk].x = bRp[(size_t)(4 * kk) * D_];
            br[kk].y = bRp[(size_t)(4 * kk + 1) * D_];
        }
    }
    const float bv = bias[n];
    const float av = alpha[n];

    const int nstrips = n_dst >> 4;
    const int r  = t >> 4;          // staging row 0..15
    const int c0 = (t & 15) * 8;    // staging col 0,8,...,120

    for (int strip = blockIdx.x; strip < nstrips; strip += gridDim.x) {
        const int m0 = strip * 16;

        // Cooperative stage: 16x128 mean tile (sum * 1/max(cnt,1)) + x tile.
        {
            const int grow = m0 + r;
            const float invc = 1.0f / fmaxf(cnt[grow], 1.0f);
            const float4* sp = (const float4*)(sum  + (size_t)grow * D_ + c0);
            const float4* xp = (const float4*)(xdst + (size_t)grow * D_ + c0);
            float4 s0 = sp[0], s1 = sp[1];
            float4 x0 = xp[0], x1 = xp[1];
            float* dm = &ldsMean[r * LDS_STRIDE + c0];
            float* dx = &ldsX[r * LDS_STRIDE + c0];
            dm[0] = s0.x * invc; dm[1] = s0.y * invc;
            dm[2] = s0.z * invc; dm[3] = s0.w * invc;
            dm[4] = s1.x * invc; dm[5] = s1.y * invc;
            dm[6] = s1.z * invc; dm[7] = s1.w * invc;
            dx[0] = x0.x; dx[1] = x0.y; dx[2] = x0.z; dx[3] = x0.w;
            dx[4] = x1.x; dx[5] = x1.y; dx[6] = x1.z; dx[7] = x1.w;
        }
        __syncthreads();

        const float* aM = &ldsMean[mn * LDS_STRIDE + 2 * half];
        const float* aX = &ldsX[mn * LDS_STRIDE + 2 * half];

        v8f acc = {0.f, 0.f, 0.f, 0.f, 0.f, 0.f, 0.f, 0.f};
#pragma unroll
        for (int kk = 0; kk < 32; ++kk) {
            v2f a0 = *(const v2f*)(aM + 4 * kk);          // mean @ Wl
            acc = __builtin_amdgcn_wmma_f32_16x16x4_f32(
                false, a0, false, bl[kk], (short)0, acc, false, false);
            v2f a1 = *(const v2f*)(aX + 4 * kk);          // x_dst @ Wr
            acc = __builtin_amdgcn_wmma_f32_16x16x4_f32(
                false, a1, false, br[kk], (short)0, acc, false, false);
        }

        // Epilogue: bias + PReLU. N is fixed per lane -> lane scalars.
        float* orow = out + (size_t)(m0 + 8 * half) * D_ + n;
#pragma unroll
        for (int v = 0; v < 8; ++v) {
            float o = acc[v] + bv;
            o = (o > 0.0f) ? o : av * o;
            orow[(size_t)v * D_] = o;
        }
        __syncthreads();   // LDS re-staged next iteration
    }
}

// ---------------------------------------------------------------------------
static void run_layer(const float* xsrc, int n_dst,
                      const int* src, const int* dst, int E,
                      const float* Wl, const float* Wr,
                      const float* b, const float* a,
                      float* sum, float* cnt, float* out, hipStream_t stream)
{
    const long nsum = (long)n_dst * D_;
    zero_f32<<<(int)((nsum + 255) / 256), 256, 0, stream>>>(sum, nsum);
    zero_f32<<<(n_dst + 255) / 256, 256, 0, stream>>>(cnt, n_dst);

    const long threads = (long)E * 32;
    scatter_accum<<<(int)((threads + 255) / 256), 256, 0, stream>>>(
        xsrc, src, dst, sum, cnt, E);

    const int nstrips = n_dst / 16;
    const int grid = (nstrips < 4096) ? nstrips : 4096;
    sage_gemm_prelu<<<grid, 256, 0, stream>>>(
        xsrc, sum, cnt, Wl, Wr, b, a, out, n_dst);
}

extern "C" void kernel_launch(void* const* d_in, const int* in_sizes, int n_in,
                              void* d_out, int out_size, void* d_ws, size_t ws_size,
                              hipStream_t stream)
{
    const float* x    = (const float*)d_in[0];
    const int*   src1 = (const int*)d_in[1];
    const int*   dst1 = (const int*)d_in[2];
    const int*   src2 = (const int*)d_in[3];
    const int*   dst2 = (const int*)d_in[4];
    const int*   src3 = (const int*)d_in[5];
    const int*   dst3 = (const int*)d_in[6];
    const float* Wl1  = (const float*)d_in[7];
    const float* Wr1  = (const float*)d_in[8];
    const float* b1   = (const float*)d_in[9];
    const float* a1   = (const float*)d_in[10];
    const float* Wl2  = (const float*)d_in[11];
    const float* Wr2  = (const float*)d_in[12];
    const float* b2   = (const float*)d_in[13];
    const float* a2   = (const float*)d_in[14];
    const float* Wl3  = (const float*)d_in[15];
    const float* Wr3  = (const float*)d_in[16];
    const float* b3   = (const float*)d_in[17];
    const float* a3   = (const float*)d_in[18];

    // Workspace layout (floats): [sum: N1*128][cnt: N1][h1: N1*128][h2: N2*128]
    float* ws  = (float*)d_ws;
    float* sum = ws;
    float* cnt = sum + (size_t)N1_ * D_;
    float* h1  = cnt + N1_;
    float* h2  = h1 + (size_t)N1_ * D_;

    run_layer(x,  N1_, src1, dst1, E1_, Wl1, Wr1, b1, a1, sum, cnt, h1, stream);
    run_layer(h1, N2_, src2, dst2, E2_, Wl2, Wr2, b2, a2, sum, cnt, h2, stream);
    run_layer(h2, N3_, src3, dst3, E3_, Wl3, Wr3, b3, a3, sum, cnt,
              (float*)d_out, stream);
}